// HeteroGATLayer_43284680409831
// MI455X (gfx1250) — compile-verified
//
#include <hip/hip_runtime.h>
#include <math.h>
#include <stdint.h>

#define NN 50000      // nodes per type
#define EE 400000     // edges per type
#define INDIM 128
#define HD 128        // H*D
#define NHEAD 4
#define DHEAD 32

typedef float v2f __attribute__((ext_vector_type(2)));
typedef float v8f __attribute__((ext_vector_type(8)));
typedef unsigned int v4u __attribute__((ext_vector_type(4)));
typedef int v4i __attribute__((ext_vector_type(4)));
typedef int v8i __attribute__((ext_vector_type(8)));

// ---------------------------------------------------------------------------
// Projection: Y[N,128] = X[N,128] @ W[128,128] + b, via V_WMMA_F32_16X16X4_F32.
// Block = 256 threads = 8 waves. Wave w computes the 16x16 tile at
// (rows blockIdx.x*16, cols w*16). K loop: 128/4 = 32 WMMA ops.
// W (64 KB) is staged once per workgroup into LDS by the Tensor Data Mover
// (TDM descriptor: 2D 128x128 tile of 4-byte elements, stride 128), so the
// B operand comes from ds loads while the vmem pipe streams the A rows.
// ---------------------------------------------------------------------------
__global__ void __launch_bounds__(256)
proj_wmma_f32(const float* __restrict__ X, const float* __restrict__ W,
              const float* __restrict__ bias, float* __restrict__ Y) {
  __shared__ float lds_w[INDIM * HD];   // 64 KB weight stage

  const int wave = threadIdx.x >> 5;
  const int lane = threadIdx.x & 31;

#if defined(__gfx1250__) && __has_builtin(__builtin_amdgcn_tensor_load_to_lds)
  if (wave == 0) {
    // D# group 0: count=1, lds_addr, global_addr[56:0], type=2 ("image")
    const unsigned lds_off = (unsigned)(uintptr_t)&lds_w[0];  // low 32b = LDS offset
    const unsigned long long ga = (unsigned long long)(uintptr_t)W;
    v4u g0;
    g0.x = 1u;                                        // count = 1 valid descriptor
    g0.y = lds_off;                                   // lds_addr (bytes)
    g0.z = (unsigned)(ga & 0xFFFFFFFFull);            // global_addr[31:0]
    g0.w = (unsigned)((ga >> 32) & 0x01FFFFFFull)     // global_addr[56:32]
           | 0x80000000u;                             // type = 2
    // D# group 1: data_size=2 (4B), tensor_dim0/1 = 128, tile_dim0/1 = 128,
    // tensor_dim0_stride = 128, everything else (mask/pad/iterate) = 0
    v8i g1;
    g1[0] = 0x00020000;          // [17:16] data_size = 2 -> 4 bytes
    g1[1] = (INDIM & 0xFFFF) << 16;                   // tensor_dim0[15:0]
    g1[2] = ((INDIM >> 16) & 0xFFFF) | ((INDIM & 0xFFFF) << 16); // dim0 hi | dim1 lo
    g1[3] = ((INDIM >> 16) & 0xFFFF) | ((HD & 0xFFFF) << 16);    // dim1 hi | tile_dim0
    g1[4] = (INDIM & 0xFFFF);                         // tile_dim1 (tile_dim2 = 0)
    g1[5] = HD;                                       // tensor_dim0_stride[31:0]
    g1[6] = 0;                                        // stride hi | dim1_stride lo
    g1[7] = 0;
    v4i gz4 = {0, 0, 0, 0};                           // groups 2/3 unused (2D)
    v8i gz8 = {0, 0, 0, 0, 0, 0, 0, 0};               // extra group (clang-23 form)
    __builtin_amdgcn_tensor_load_to_lds(g0, g1, gz4, gz4, gz8, 0);
    __builtin_amdgcn_s_wait_tensorcnt(0);
  }
  __syncthreads();
#else
  for (int i = threadIdx.x; i < INDIM * HD; i += 256) lds_w[i] = W[i];
  __syncthreads();
#endif

  const int m0 = blockIdx.x << 4;
  const int n0 = wave << 4;
  const int lr = lane & 15;
  const int hi = lane >> 4;

  const float* xrow = X + (size_t)(m0 + lr) * INDIM;
  const float* wcol = lds_w + (n0 + lr);

  v8f c = {0.f, 0.f, 0.f, 0.f, 0.f, 0.f, 0.f, 0.f};
#pragma unroll 4
  for (int kb = 0; kb < INDIM; kb += 4) {
    const int k = kb + (hi << 1);
    v2f a;
    a.x = xrow[k];
    a.y = xrow[k + 1];
    v2f b;
    b.x = wcol[k * HD];
    b.y = wcol[(k + 1) * HD];
    c = __builtin_amdgcn_wmma_f32_16x16x4_f32(false, a, false, b, (short)0, c,
                                              false, false);
  }
  const float bv = bias[n0 + lr];
#pragma unroll
  for (int r = 0; r < 8; ++r) {
    const int row = m0 + r + (hi << 3);
    Y[(size_t)row * HD + n0 + lr] = c[r] + bv;
  }
}

// ---------------------------------------------------------------------------
// Attention logits: a[n,h] = sum_d Wh[n,h,d] * p[h,d]   (one thread per (n,h))
// ---------------------------------------------------------------------------
__global__ void attn_logits(const float* __restrict__ Wh,
                            const float* __restrict__ p, float* __restrict__ a) {
  const int t = blockIdx.x * blockDim.x + threadIdx.x;
  if (t >= NN * NHEAD) return;
  const int n = t >> 2, h = t & 3;
  const float* row = Wh + (size_t)n * HD + h * DHEAD;
  const float* pp = p + h * DHEAD;
  float s = 0.f;
#pragma unroll
  for (int d = 0; d < DHEAD; ++d) s += row[d] * pp[d];
  a[t] = s;
}

__global__ void fill_f32(float* __restrict__ p, float v, int n) {
  const int i = blockIdx.x * blockDim.x + threadIdx.x;
  if (i < n) p[i] = v;
}

__device__ __forceinline__ float lrelu(float x) {
  return x > 0.f ? x : 0.2f * x;
}

// float atomic max via monotone signed/unsigned int encoding (init to -inf)
__device__ __forceinline__ void atomicMaxF(float* addr, float v) {
  if (v >= 0.f)
    atomicMax((int*)addr, __float_as_int(v));
  else
    atomicMin((unsigned int*)addr, __float_as_uint(v));
}

__global__ void edge_max(const int* __restrict__ src, const int* __restrict__ dst,
                         const float* __restrict__ as, const float* __restrict__ ad,
                         float* __restrict__ m) {
  const int t = blockIdx.x * blockDim.x + threadIdx.x;
  if (t >= EE * NHEAD) return;
  const int e = t >> 2, h = t & 3;
  const int s = src[e], d = dst[e];
  const float ev = lrelu(as[s * 4 + h] + ad[d * 4 + h]);
  atomicMaxF(&m[d * 4 + h], ev);
}

__global__ void edge_expsum(const int* __restrict__ src, const int* __restrict__ dst,
                            const float* __restrict__ as, const float* __restrict__ ad,
                            const float* __restrict__ m, float* __restrict__ ssum) {
  const int t = blockIdx.x * blockDim.x + threadIdx.x;
  if (t >= EE * NHEAD) return;
  const int e = t >> 2, h = t & 3;
  const int s = src[e], d = dst[e];
  const float ev = lrelu(as[s * 4 + h] + ad[d * 4 + h]);
  atomicAdd(&ssum[d * 4 + h], __expf(ev - m[d * 4 + h]));
}

// One wave per edge; lane covers (head = lane>>3, 4 consecutive d's).
__global__ void __launch_bounds__(256)
scatter_attn(const float* __restrict__ Wh, const int* __restrict__ src,
             const int* __restrict__ dst, const float* __restrict__ as,
             const float* __restrict__ ad, const float* __restrict__ m,
             const float* __restrict__ ssum, float* __restrict__ out) {
  const int e = blockIdx.x * (blockDim.x >> 5) + (threadIdx.x >> 5);
  if (e >= EE) return;
  const int lane = threadIdx.x & 31;
  const int h = lane >> 3;
  const int s = src[e], d = dst[e];
  const float ev = lrelu(as[s * 4 + h] + ad[d * 4 + h]);
  const float coef = __expf(ev - m[d * 4 + h]) / ssum[d * 4 + h];
  const float4 v = ((const float4*)(Wh + (size_t)s * HD))[lane];
  float* pd = out + (size_t)d * HD + lane * 4;
  atomicAdd(pd + 0, v.x * coef);
  atomicAdd(pd + 1, v.y * coef);
  atomicAdd(pd + 2, v.z * coef);
  atomicAdd(pd + 3, v.w * coef);
}

__global__ void __launch_bounds__(256)
scatter_plain(const float* __restrict__ Wh, const int* __restrict__ src,
              const int* __restrict__ dst, float* __restrict__ out) {
  const int e = blockIdx.x * (blockDim.x >> 5) + (threadIdx.x >> 5);
  if (e >= EE) return;
  const int lane = threadIdx.x & 31;
  const int s = src[e], d = dst[e];
  const float4 v = ((const float4*)(Wh + (size_t)s * HD))[lane];
  float* pd = out + (size_t)d * HD + lane * 4;
  atomicAdd(pd + 0, v.x);
  atomicAdd(pd + 1, v.y);
  atomicAdd(pd + 2, v.z);
  atomicAdd(pd + 3, v.w);
}

__global__ void relu_inplace(float* __restrict__ p, int n) {
  const int i = blockIdx.x * blockDim.x + threadIdx.x;
  if (i < n) {
    const float v = p[i];
    p[i] = v > 0.f ? v : 0.f;
  }
}

// ---------------------------------------------------------------------------
extern "C" void kernel_launch(void* const* d_in, const int* in_sizes, int n_in,
                              void* d_out, int out_size, void* d_ws,
                              size_t ws_size, hipStream_t stream) {
  // inputs in setup_inputs() order
  const float* feat_crew  = (const float*)d_in[0];
  const float* feat_plane = (const float*)d_in[1];
  const float* feat_state = (const float*)d_in[2];
  const float* feat_value = (const float*)d_in[3];
  // PROJS order: crew, plane, c_in, repairing, p_in, p_to, repaired_by, s_in, s_to, v_to
  const float* W_crew = (const float*)d_in[4];        const float* b_crew = (const float*)d_in[5];
  const float* W_plane = (const float*)d_in[6];       const float* b_plane = (const float*)d_in[7];
  const float* W_c_in = (const float*)d_in[8];        const float* b_c_in = (const float*)d_in[9];
  const float* W_repairing = (const float*)d_in[10];  const float* b_repairing = (const float*)d_in[11];
  const float* W_p_in = (const float*)d_in[12];       const float* b_p_in = (const float*)d_in[13];
  const float* W_p_to = (const float*)d_in[14];       const float* b_p_to = (const float*)d_in[15];
  const float* W_repaired_by = (const float*)d_in[16];const float* b_repaired_by = (const float*)d_in[17];
  const float* W_s_in = (const float*)d_in[18];       const float* b_s_in = (const float*)d_in[19];
  const float* W_s_to = (const float*)d_in[20];       const float* b_s_to = (const float*)d_in[21];
  const float* W_v_to = (const float*)d_in[22];       const float* b_v_to = (const float*)d_in[23];
  const float* c_in_src_p = (const float*)d_in[24];
  const float* c_in_dst_p = (const float*)d_in[25];
  const float* p_in_src_p = (const float*)d_in[26];
  const float* p_in_dst_p = (const float*)d_in[27];
  // ETYPES order: c_in, repairing, p_in, p_to, repaired_by, s_in, s_to, v_to
  const int* src_c_in = (const int*)d_in[28];        const int* dst_c_in = (const int*)d_in[29];
  const int* src_repairing = (const int*)d_in[30];   const int* dst_repairing = (const int*)d_in[31];
  const int* src_p_in = (const int*)d_in[32];        const int* dst_p_in = (const int*)d_in[33];
  const int* src_p_to = (const int*)d_in[34];        const int* dst_p_to = (const int*)d_in[35];
  const int* src_repaired_by = (const int*)d_in[36]; const int* dst_repaired_by = (const int*)d_in[37];
  // src/dst s_in (38,39) unused by the reference combine
  const int* src_s_to = (const int*)d_in[40];        const int* dst_s_to = (const int*)d_in[41];
  // src/dst v_to (42,43) unused (v_to projection feeds h_value base directly)

  float* out = (float*)d_out;           // [4, N, H, D]
  float* out_crew  = out + (size_t)0 * NN * HD;
  float* out_plane = out + (size_t)1 * NN * HD;
  float* out_state = out + (size_t)2 * NN * HD;
  float* out_value = out + (size_t)3 * NN * HD;

  // workspace layout (floats): one reusable Wh buffer + attention scalars
  float* ws = (float*)d_ws;
  float* wh      = ws;                                  // N*128
  float* asrc_c  = wh + (size_t)NN * HD;                // N*H each below
  float* adst_c  = asrc_c + NN * NHEAD;
  float* m_c     = adst_c + NN * NHEAD;
  float* s_c     = m_c + NN * NHEAD;
  float* asrc_p  = s_c + NN * NHEAD;
  float* adst_p  = asrc_p + NN * NHEAD;
  float* m_p     = adst_p + NN * NHEAD;
  float* s_p     = m_p + NN * NHEAD;

  const dim3 projGrid(NN / 16);          // 3125 blocks, 8 waves each
  const dim3 blk256(256);
  const int nhGrid = (NN * NHEAD + 255) / 256;
  const int ehGrid = (EE * NHEAD + 255) / 256;
  const int edgeGrid = (EE + 7) / 8;     // wave-per-edge, 8 edges/block
  const int outElems = 4 * NN * HD;
  const float ninf = -INFINITY;

  // base projections straight into output slots
  proj_wmma_f32<<<projGrid, blk256, 0, stream>>>(feat_crew, W_crew, b_crew, out_crew);
  proj_wmma_f32<<<projGrid, blk256, 0, stream>>>(feat_plane, W_plane, b_plane, out_plane);
  proj_wmma_f32<<<projGrid, blk256, 0, stream>>>(feat_state, W_s_in, b_s_in, out_state);
  proj_wmma_f32<<<projGrid, blk256, 0, stream>>>(feat_value, W_v_to, b_v_to, out_value);

  // dst attention logits from Wh_s_in (must precede scatters into out_state)
  attn_logits<<<nhGrid, blk256, 0, stream>>>(out_state, c_in_dst_p, adst_c);
  attn_logits<<<nhGrid, blk256, 0, stream>>>(out_state, p_in_dst_p, adst_p);

  // per-call init of softmax accumulators
  fill_f32<<<nhGrid, blk256, 0, stream>>>(m_c, ninf, NN * NHEAD);
  fill_f32<<<nhGrid, blk256, 0, stream>>>(s_c, 0.f, NN * NHEAD);
  fill_f32<<<nhGrid, blk256, 0, stream>>>(m_p, ninf, NN * NHEAD);
  fill_f32<<<nhGrid, blk256, 0, stream>>>(s_p, 0.f, NN * NHEAD);

  // relation c_in: crew -> state (GAT)
  proj_wmma_f32<<<projGrid, blk256, 0, stream>>>(feat_crew, W_c_in, b_c_in, wh);
  attn_logits<<<nhGrid, blk256, 0, stream>>>(wh, c_in_src_p, asrc_c);
  edge_max<<<ehGrid, blk256, 0, stream>>>(src_c_in, dst_c_in, asrc_c, adst_c, m_c);
  edge_expsum<<<ehGrid, blk256, 0, stream>>>(src_c_in, dst_c_in, asrc_c, adst_c, m_c, s_c);
  scatter_attn<<<edgeGrid, blk256, 0, stream>>>(wh, src_c_in, dst_c_in, asrc_c,
                                                adst_c, m_c, s_c, out_state);

  // relation p_in: plane -> state (GAT)
  proj_wmma_f32<<<projGrid, blk256, 0, stream>>>(feat_plane, W_p_in, b_p_in, wh);
  attn_logits<<<nhGrid, blk256, 0, stream>>>(wh, p_in_src_p, asrc_p);
  edge_max<<<ehGrid, blk256, 0, stream>>>(src_p_in, dst_p_in, asrc_p, adst_p, m_p);
  edge_expsum<<<ehGrid, blk256, 0, stream>>>(src_p_in, dst_p_in, asrc_p, adst_p, m_p, s_p);
  scatter_attn<<<edgeGrid, blk256, 0, stream>>>(wh, src_p_in, dst_p_in, asrc_p,
                                                adst_p, m_p, s_p, out_state);

  // plain copy_src + sum relations
  proj_wmma_f32<<<projGrid, blk256, 0, stream>>>(feat_plane, W_repaired_by, b_repaired_by, wh);
  scatter_plain<<<edgeGrid, blk256, 0, stream>>>(wh, src_repaired_by, dst_repaired_by, out_crew);

  proj_wmma_f32<<<projGrid, blk256, 0, stream>>>(feat_crew, W_repairing, b_repairing, wh);
  scatter_plain<<<edgeGrid, blk256, 0, stream>>>(wh, src_repairing, dst_repairing, out_plane);

  proj_wmma_f32<<<projGrid, blk256, 0, stream>>>(feat_plane, W_p_to, b_p_to, wh);
  scatter_plain<<<edgeGrid, blk256, 0, stream>>>(wh, src_p_to, dst_p_to, out_value);

  proj_wmma_f32<<<projGrid, blk256, 0, stream>>>(feat_state, W_s_to, b_s_to, wh);
  scatter_plain<<<edgeGrid, blk256, 0, stream>>>(wh, src_s_to, dst_s_to, out_value);

  // final ReLU over the whole stacked output
  relu_inplace<<<(outElems + 255) / 256, blk256, 0, stream>>>(out, outElems);
}